// FlatGNN_73856257622089
// MI455X (gfx1250) — compile-verified
//
#include <hip/hip_runtime.h>

// FlatGNN for MI455X (gfx1250): fp32 WMMA (v_wmma_f32_16x16x4_f32) GEMMs with
// LDS-staged weight panels (async global->LDS copies via the gfx1250
// GLOBAL_LOAD_ASYNC_TO_LDS path), 2x2 register blocking per wave, and
// VALU + global atomics for the irregular SPMM hops.

typedef __attribute__((ext_vector_type(2))) float v2f;
typedef __attribute__((ext_vector_type(8))) float v8f;
typedef int v4i_vs __attribute__((vector_size(16)));   // builtin's element type
typedef __attribute__((address_space(1))) v4i_vs* gptr_b128;
typedef __attribute__((address_space(3))) v4i_vs* lptr_b128;

#define IND 128    // input feature dim
#define HD  256    // hidden dim
#define TPB 256    // 8 waves per block (wave32)
#define SLABW 32   // output-column slab per block
#define KC 64      // K-panel depth staged in LDS
#define BS_STRIDE 40  // padded LDS row stride (words): 2 k-rows apart -> +16 banks

#if defined(__has_builtin)
#if __has_builtin(__builtin_amdgcn_global_load_async_to_lds_b128) && \
    __has_builtin(__builtin_amdgcn_s_wait_asynccnt)
#define USE_ASYNC_LDS 1
#endif
#endif
#ifndef USE_ASYNC_LDS
#define USE_ASYNC_LDS 0
#endif

#define WMMA4(a, b, c)                                                        \
  __builtin_amdgcn_wmma_f32_16x16x4_f32(false, (a), false, (b), (short)0, (c), \
                                        false, false)

// ---------------- adjacency normalization ----------------

__global__ void fgnn_fill1(float* __restrict__ p, int n) {
  int i = blockIdx.x * blockDim.x + threadIdx.x;
  if (i < n) p[i] = 1.0f;   // self-loop contributes 1 to each degree
}

__global__ void fgnn_degree(const int* __restrict__ row, float* __restrict__ deg,
                            long long e) {
  long long i = (long long)blockIdx.x * blockDim.x + threadIdx.x;
  if (i < e) atomicAdd(&deg[row[i]], 1.0f);
}

__global__ void fgnn_dinv(float* __restrict__ d, int n) {
  int i = blockIdx.x * blockDim.x + threadIdx.x;
  if (i < n) d[i] = rsqrtf(d[i]);   // deg >= 1 always (self-loop)
}

// y[i,:] = dinv[i]^2 * x[i,:]  (self-loop term of A_hat @ x)
__global__ void fgnn_selfloop(float* __restrict__ y, const float* __restrict__ x,
                              const float* __restrict__ dinv, int n) {
  int i = blockIdx.x * blockDim.x + threadIdx.x;
  if (i < n * IND) {
    int node = i >> 7;               // IND == 128
    float w = dinv[node];
    y[i] = w * w * x[i];
  }
}

// edge term: y[r,:] += dinv[r]*dinv[c] * x[c,:]; 32 threads/edge, 4 floats each
__global__ void fgnn_spmm(const int* __restrict__ row, const int* __restrict__ col,
                          const float* __restrict__ dinv,
                          const float* __restrict__ x, float* __restrict__ y,
                          long long e) {
  long long idx = (long long)blockIdx.x * blockDim.x + threadIdx.x;
  if (idx >= e * 32) return;
  long long ed = idx >> 5;
  int f0 = (int)(idx & 31) << 2;
  int r = row[ed];
  int c = col[ed];
  float w = dinv[r] * dinv[c];
  const float4 v = *(const float4*)(x + (size_t)c * IND + f0);
  float* yp = y + (size_t)r * IND + f0;
  atomicAdd(yp + 0, w * v.x);
  atomicAdd(yp + 1, w * v.y);
  atomicAdd(yp + 2, w * v.z);
  atomicAdd(yp + 3, w * v.w);
}

// ---------------- LDS staging of a KC x SLABW weight panel ----------------
// gsrc points at W[k0][colBase] (row stride HD). Each of 256 threads moves
// 2 x 16B. Async path uses gfx1250 GLOBAL_LOAD_ASYNC_TO_LDS (ASYNCcnt).
// Generic->AS3 cast via uintptr_t: low 32 bits of a generic shared address
// are the LDS offset (ISA aperture layout).

__device__ __forceinline__ void fgnn_stage_b(const float* __restrict__ gsrc,
                                             float* __restrict__ dst, int tid) {
  const int r  = tid >> 3;           // 0..31 (k-row within panel)
  const int c4 = (tid & 7) << 2;     // 0,4,...,28 (column quad)
  const float* s0 = gsrc + (size_t)r * HD + c4;
  const float* s1 = s0 + (size_t)32 * HD;
  float* d0 = dst + r * BS_STRIDE + c4;
  float* d1 = dst + (r + 32) * BS_STRIDE + c4;
#if USE_ASYNC_LDS
  __builtin_amdgcn_global_load_async_to_lds_b128(
      (gptr_b128)(uintptr_t)s0, (lptr_b128)(uintptr_t)d0, 0, 0);
  __builtin_amdgcn_global_load_async_to_lds_b128(
      (gptr_b128)(uintptr_t)s1, (lptr_b128)(uintptr_t)d1, 0, 0);
#else
  *(float4*)d0 = *(const float4*)s0;
  *(float4*)d1 = *(const float4*)s1;
#endif
}

__device__ __forceinline__ void fgnn_wait_async() {
#if USE_ASYNC_LDS
  __builtin_amdgcn_s_wait_asynccnt(0);
#endif
}

// ---------------- fp32 WMMA compute: one KC-deep panel, 2x2 tiles ----------
// ap  : A rows [row0+m],    at k-pair 2*half of this panel (row stride = lda)
// ap2 : A rows [row0+16+m]
// bp  : LDS panel at word (2*half)*BS_STRIDE + m
__device__ __forceinline__ void fgnn_chunk(const float* __restrict__ ap,
                                           const float* __restrict__ ap2,
                                           const float* __restrict__ bp,
                                           v8f& acc0, v8f& acc1, v8f& acc2,
                                           v8f& acc3) {
#pragma unroll
  for (int k = 0; k < KC; k += 4) {
    v2f a0; a0[0] = ap[0];  a0[1] = ap[1];
    v2f a1; a1[0] = ap2[0]; a1[1] = ap2[1];
    v2f b0; b0[0] = bp[0];  b0[1] = bp[BS_STRIDE];
    v2f b1; b1[0] = bp[16]; b1[1] = bp[BS_STRIDE + 16];
    acc0 = WMMA4(a0, b0, acc0);
    acc1 = WMMA4(a0, b1, acc1);
    acc2 = WMMA4(a1, b0, acc2);
    acc3 = WMMA4(a1, b1, acc3);
    ap += 4;
    ap2 += 4;
    bp += 4 * BS_STRIDE;
  }
}

// y0 pre-offset to Y[row0 + 8*half][colOut + m]; bcol pre-offset bias+col+m.
__device__ __forceinline__ void fgnn_store4(float* __restrict__ y0, size_t ldy,
                                            const float* __restrict__ bcol,
                                            int relu, const v8f& acc0,
                                            const v8f& acc1, const v8f& acc2,
                                            const v8f& acc3) {
  const float bv0 = bcol[0];
  const float bv1 = bcol[16];
  float* y1 = y0 + (size_t)16 * ldy;
#pragma unroll
  for (int v = 0; v < 8; ++v) {
    float v00 = acc0[v] + bv0, v01 = acc1[v] + bv1;
    float v10 = acc2[v] + bv0, v11 = acc3[v] + bv1;
    if (relu) {
      v00 = fmaxf(v00, 0.0f); v01 = fmaxf(v01, 0.0f);
      v10 = fmaxf(v10, 0.0f); v11 = fmaxf(v11, 0.0f);
    }
    y0[(size_t)v * ldy] = v00;
    y0[(size_t)v * ldy + 16] = v01;
    y1[(size_t)v * ldy] = v10;
    y1[(size_t)v * ldy + 16] = v11;
  }
}

// ---------------- Y[N x 256] = act(A[N x kdim] @ W[kdim x 256] + bias) ------
// grid: x = 32-row wave-tiles / 8, y = column slab (HD/SLABW)

__global__ void fgnn_gemm(const float* __restrict__ A, int kdim,
                          const float* __restrict__ W, const float* __restrict__ bias,
                          float* __restrict__ Y, int relu, int nrows) {
  __shared__ float Bs[2][KC * BS_STRIDE];
  const int tid = threadIdx.x;
  const int wave = tid >> 5, lane = tid & 31;
  const int half = lane >> 4, m = lane & 15;
  const int colBase = blockIdx.y * SLABW;
  const int rowTiles = nrows >> 5;                 // 32-row tiles
  int rt = blockIdx.x * (blockDim.x >> 5) + wave;
  const bool active = rt < rowTiles;
  if (!active) rt = rowTiles - 1;                  // clamp: keep wave for barriers
  const int row0 = rt << 5;

  const int nchunks = kdim / KC;
  fgnn_stage_b(W + colBase, Bs[0], tid);
  fgnn_wait_async();
  __syncthreads();

  v8f acc0 = {}, acc1 = {}, acc2 = {}, acc3 = {};
  const float* aBase = A + (size_t)(row0 + m) * kdim + 2 * half;
  for (int c = 0; c < nchunks; ++c) {
    if (c + 1 < nchunks)
      fgnn_stage_b(W + (size_t)(c + 1) * KC * HD + colBase, Bs[(c + 1) & 1], tid);
    fgnn_chunk(aBase + c * KC, aBase + (size_t)16 * kdim + c * KC,
               Bs[c & 1] + (2 * half) * BS_STRIDE + m, acc0, acc1, acc2, acc3);
    fgnn_wait_async();
    __syncthreads();
  }

  if (active) {
    fgnn_store4(Y + (size_t)(row0 + 8 * half) * HD + colBase + m, HD,
                bias + colBase + m, relu, acc0, acc1, acc2, acc3);
  }
}

// ---- out[:, relBase:relBase+256] = relu( sum_s h_s @ Wr_s + br ) -----------
// Wr rows are contiguous across the nseg K=256 segments, so the LDS panel
// stream is linear in t; only the A base pointer switches per segment.

__global__ void fgnn_rel(const float* __restrict__ h, size_t hopStride, int nseg,
                         const float* __restrict__ Wr, const float* __restrict__ br,
                         float* __restrict__ out, int outld, int relBase, int nrows) {
  __shared__ float Bs[2][KC * BS_STRIDE];
  const int tid = threadIdx.x;
  const int wave = tid >> 5, lane = tid & 31;
  const int half = lane >> 4, m = lane & 15;
  const int colBase = blockIdx.y * SLABW;
  const int rowTiles = nrows >> 5;
  int rt = blockIdx.x * (blockDim.x >> 5) + wave;
  const bool active = rt < rowTiles;
  if (!active) rt = rowTiles - 1;
  const int row0 = rt << 5;

  const int chunksPerSeg = HD / KC;                // 4
  const int nchunks = nseg * chunksPerSeg;         // 12
  fgnn_stage_b(Wr + colBase, Bs[0], tid);
  fgnn_wait_async();
  __syncthreads();

  v8f acc0 = {}, acc1 = {}, acc2 = {}, acc3 = {};
  for (int t = 0; t < nchunks; ++t) {
    if (t + 1 < nchunks)
      fgnn_stage_b(Wr + (size_t)(t + 1) * KC * HD + colBase, Bs[(t + 1) & 1], tid);
    const int seg = t / chunksPerSeg;
    const int kc = t - seg * chunksPerSeg;
    const float* aBase = h + (size_t)seg * hopStride +
                         (size_t)(row0 + m) * HD + 2 * half + kc * KC;
    fgnn_chunk(aBase, aBase + (size_t)16 * HD,
               Bs[t & 1] + (2 * half) * BS_STRIDE + m, acc0, acc1, acc2, acc3);
    fgnn_wait_async();
    __syncthreads();
  }

  if (active) {
    fgnn_store4(out + (size_t)(row0 + 8 * half) * outld + relBase + colBase + m,
                outld, br + colBase + m, /*relu=*/1, acc0, acc1, acc2, acc3);
  }
}

// ---------------- host orchestration ----------------

extern "C" void kernel_launch(void* const* d_in, const int* in_sizes, int n_in,
                              void* d_out, int out_size, void* d_ws, size_t ws_size,
                              hipStream_t stream) {
  const float* features = (const float*)d_in[0];
  const float* W1 = (const float*)d_in[1];
  const float* b1 = (const float*)d_in[2];
  const float* W2 = (const float*)d_in[3];
  const float* b2 = (const float*)d_in[4];
  const float* Wr = (const float*)d_in[5];
  const float* br = (const float*)d_in[6];
  const int* row = (const int*)d_in[7];
  const int* col = (const int*)d_in[8];

  const int N = in_sizes[0] / IND;             // 100000 (divisible by 32)
  const long long E = in_sizes[7];             // 1600000
  const int K = in_sizes[2] / HD;              // 7 hops incl. identity
  const int R = in_sizes[6] / HD;              // 5 relations
  const int D = in_sizes[5] / (R * HD);        // 768 = INTERVAL*H
  const int NSEG = D / HD;                     // 3
  const int OUTLD = R * HD;                    // 1280

  // workspace carve-out (256B aligned slices)
  char* p = (char*)d_ws;
  auto take = [&](size_t bytes) -> float* {
    char* q = p;
    p += (bytes + 255) & ~(size_t)255;
    return (float*)q;
  };
  float* dinv = take((size_t)N * 4);
  float* xA   = take((size_t)N * IND * 4);
  float* xB   = take((size_t)N * IND * 4);
  float* tmp  = take((size_t)N * HD * 4);
  float* h    = take((size_t)K * N * HD * 4);

  // D^{-1/2} with self-loops
  fgnn_fill1<<<(N + TPB - 1) / TPB, TPB, 0, stream>>>(dinv, N);
  fgnn_degree<<<(unsigned)((E + TPB - 1) / TPB), TPB, 0, stream>>>(row, dinv, E);
  fgnn_dinv<<<(N + TPB - 1) / TPB, TPB, 0, stream>>>(dinv, N);

  const int rowTiles32 = N / 32;                       // 3125
  dim3 gemmGrid((rowTiles32 + 7) / 8, HD / SLABW);     // (391, 8)

  // hop loop: MLP(hop_k) -> h[k];  SPMM -> hop_{k+1}
  const float* xk = features;
  for (int k = 0; k < K; ++k) {
    fgnn_gemm<<<gemmGrid, TPB, 0, stream>>>(
        xk, IND, W1 + (size_t)k * IND * HD, b1 + (size_t)k * HD, tmp, /*relu=*/1, N);
    fgnn_gemm<<<gemmGrid, TPB, 0, stream>>>(
        tmp, HD, W2 + (size_t)k * HD * HD, b2 + (size_t)k * HD,
        h + (size_t)k * N * HD, /*relu=*/0, N);
    if (k + 1 < K) {
      float* xn = (k % 2 == 0) ? xA : xB;
      fgnn_selfloop<<<(N * IND + TPB - 1) / TPB, TPB, 0, stream>>>(xn, xk, dinv, N);
      fgnn_spmm<<<(unsigned)((E * 32 + TPB - 1) / TPB), TPB, 0, stream>>>(
          row, col, dinv, xk, xn, E);
      xk = xn;
    }
  }

  // relation GEMMs: window r = hops [r, r+1, r+2] -> out[:, r*256:(r+1)*256]
  for (int r = 0; r < R; ++r) {
    fgnn_rel<<<gemmGrid, TPB, 0, stream>>>(
        h + (size_t)r * N * HD, (size_t)N * HD, NSEG,
        Wr + (size_t)r * D * HD, br + (size_t)r * HD,
        (float*)d_out, OUTLD, r * HD, N);
  }
}